// ResLayer_85555748536419
// MI455X (gfx1250) — compile-verified
//
#include <hip/hip_runtime.h>
#include <hip/hip_bf16.h>

#define N_NODES 50000
#define N_EDGES 400000
#define E_TOT   (N_EDGES + N_NODES)   // edges + self loops
#define HEADS   6
#define DIMS    32
#define FEAT    192                    // HEADS*DIMS
#define NF      (N_NODES * FEAT)
#define NH      (N_NODES * HEADS)
#define NEG_SLOPE 0.2f

typedef float v2f __attribute__((ext_vector_type(2)));
typedef float v8f __attribute__((ext_vector_type(8)));

// ---------------------------------------------------------------------------
// float atomic max via int/uint ordering trick (amax initialized to -inf)
// ---------------------------------------------------------------------------
__device__ inline void atomicMaxFloat(float* addr, float val) {
    int iv = __float_as_int(val);
    if (iv >= 0)
        atomicMax((int*)addr, iv);
    else
        atomicMin((unsigned int*)addr, (unsigned int)iv);
}

// ---------------------------------------------------------------------------
// fill buffer with constant (used for -inf init of segment max)
// ---------------------------------------------------------------------------
__global__ void fill_kernel(float* __restrict__ p, int n, float v) {
    int i = blockIdx.x * blockDim.x + threadIdx.x;
    if (i < n) p[i] = v;
}

// ---------------------------------------------------------------------------
// per-dst edge count and edge-weight sum (for self-loop fill_value='mean')
// ---------------------------------------------------------------------------
__global__ void deg_kernel(const int* __restrict__ ei, const float* __restrict__ ew,
                           float* __restrict__ wsum, float* __restrict__ cnt) {
    int e = blockIdx.x * blockDim.x + threadIdx.x;
    if (e >= N_EDGES) return;
    int d = ei[N_EDGES + e];
    atomicAdd(&cnt[d], 1.0f);
    atomicAdd(&wsum[d], ew[e]);
}

// ---------------------------------------------------------------------------
// x_l = x@W_l + b_l ; x_r = x@W_r + b_r   via V_WMMA_F32_16X16X4_F32
// grid.x = N_NODES/16 = 3125 blocks, 256 threads = 8 waves.
// Block computes a 16-row tile of BOTH outputs (24 16x16 col-tiles total,
// 3 per wave). A-tile staged transposed in LDS: As[k*16 + r] = x[row0+r][k]
// -> fragment reads are bank-conflict-free.
// ---------------------------------------------------------------------------
__global__ __launch_bounds__(256) void gemm_xl_xr_kernel(
        const float* __restrict__ x,
        const float* __restrict__ Wl, const float* __restrict__ bl,
        const float* __restrict__ Wr, const float* __restrict__ br,
        float* __restrict__ xl, float* __restrict__ xr)
{
    __shared__ float As[FEAT * 16];           // 12 KB, k-major transposed A tile

    const int tid  = threadIdx.x;
    const int wave = tid >> 5;
    const int lane = tid & 31;
    const int half = lane >> 4;               // 0: lanes 0-15, 1: lanes 16-31
    const int l16  = lane & 15;
    const int row0 = blockIdx.x * 16;

    // cooperative, coalesced load of the 16x192 A tile (stored transposed)
    for (int i = tid; i < 16 * FEAT; i += 256) {
        int r = i / FEAT;
        int k = i - r * FEAT;
        As[k * 16 + r] = x[(size_t)(row0 + r) * FEAT + k];
    }
    __syncthreads();

    #pragma unroll
    for (int j = 0; j < 3; ++j) {
        const int p    = wave + 8 * j;        // 0..23  (wave-uniform)
        const int mat  = p / 12;              // 0 = lin_l, 1 = lin_r
        const int col0 = (p % 12) * 16;
        const float* __restrict__ W  = mat ? Wr : Wl;
        const float* __restrict__ bv = mat ? br : bl;
        float*       __restrict__ O  = mat ? xr : xl;

        v8f acc = {};
        for (int k0 = 0; k0 < FEAT; k0 += 4) {
            const int ka = k0 + 2 * half;     // A: VGPR0/1 = K {0,1} (lo lanes), {2,3} (hi)
            v2f a, b;
            a[0] = As[ka * 16 + l16];
            a[1] = As[(ka + 1) * 16 + l16];
            b[0] = W[(size_t)ka * FEAT + col0 + l16];        // B mirrors A layout
            b[1] = W[(size_t)(ka + 1) * FEAT + col0 + l16];
            acc = __builtin_amdgcn_wmma_f32_16x16x4_f32(
                      /*neg_a=*/false, a, /*neg_b=*/false, b,
                      /*c_mod=*/(short)0, acc,
                      /*reuse_a=*/false, /*reuse_b=*/false);
        }

        const float bb = bv[col0 + l16];
        #pragma unroll
        for (int v = 0; v < 8; ++v) {
            const int row = row0 + v + 8 * half;   // C/D: VGPR v -> M=v / M=v+8
            O[(size_t)row * FEAT + col0 + l16] = acc[v] + bb;
        }
    }
}

// ---------------------------------------------------------------------------
// per-edge GATv2 score: alpha[e][h] = att[h] . leaky_relu(xl[src]+xr[dst]+e_emb)
// one wave32 per edge; xor-shuffle reduction; atomic segment-max into amax.
// ---------------------------------------------------------------------------
__global__ __launch_bounds__(256) void alpha_kernel(
        const int* __restrict__ ei, const float* __restrict__ ew,
        const float* __restrict__ xl, const float* __restrict__ xr,
        const float* __restrict__ We, const float* __restrict__ att,
        const float* __restrict__ wsum, const float* __restrict__ cnt,
        float* __restrict__ alpha, float* __restrict__ amax)
{
    const int e = blockIdx.x * 8 + (threadIdx.x >> 5);
    if (e >= E_TOT) return;
    const int lane = threadIdx.x & 31;

    int s, d; float w;
    if (e < N_EDGES) {
        s = ei[e]; d = ei[N_EDGES + e]; w = ew[e];
    } else {                                   // self loop, attr = mean of incoming
        s = e - N_EDGES; d = s;
        w = wsum[s] / fmaxf(cnt[s], 1.0f);
    }

    const float* __restrict__ pl = xl + (size_t)s * FEAT;
    const float* __restrict__ pr = xr + (size_t)d * FEAT;

    #pragma unroll
    for (int h = 0; h < HEADS; ++h) {
        const int idx = h * DIMS + lane;
        float m = pl[idx] + pr[idx] + w * We[idx];
        m = (m > 0.0f) ? m : NEG_SLOPE * m;
        float p = m * att[idx];
        #pragma unroll
        for (int off = 16; off > 0; off >>= 1)
            p += __shfl_xor(p, off, 32);
        if (lane == h) {
            alpha[(size_t)e * HEADS + h] = p;
            atomicMaxFloat(&amax[(size_t)d * HEADS + h], p);
        }
    }
}

// ---------------------------------------------------------------------------
// ex = exp(alpha - amax[dst]); denom[dst] += ex  (alpha overwritten with ex)
// ---------------------------------------------------------------------------
__global__ void exp_kernel(const int* __restrict__ ei,
                           float* __restrict__ alpha,
                           const float* __restrict__ amax,
                           float* __restrict__ denom)
{
    const int i = blockIdx.x * blockDim.x + threadIdx.x;
    if (i >= E_TOT * HEADS) return;
    const int e = i / HEADS;
    const int h = i - e * HEADS;
    const int d = (e < N_EDGES) ? ei[N_EDGES + e] : (e - N_EDGES);
    const float ex = __expf(alpha[i] - amax[(size_t)d * HEADS + h]);
    alpha[i] = ex;
    atomicAdd(&denom[(size_t)d * HEADS + h], ex);
}

// ---------------------------------------------------------------------------
// out[dst] += xl[src] * (ex / denom[dst])   one wave per edge, lane = channel
// ---------------------------------------------------------------------------
__global__ __launch_bounds__(256) void agg_kernel(
        const int* __restrict__ ei,
        const float* __restrict__ xl,
        const float* __restrict__ alpha,
        const float* __restrict__ denom,
        float* __restrict__ out)
{
    const int e = blockIdx.x * 8 + (threadIdx.x >> 5);
    if (e >= E_TOT) return;
    const int lane = threadIdx.x & 31;

    int s, d;
    if (e < N_EDGES) { s = ei[e]; d = ei[N_EDGES + e]; }
    else             { s = e - N_EDGES; d = s; }

    const float* __restrict__ pl = xl + (size_t)s * FEAT;
    float* __restrict__ po = out + (size_t)d * FEAT;

    #pragma unroll
    for (int h = 0; h < HEADS; ++h) {
        const float ah = alpha[(size_t)e * HEADS + h] /
                         denom[(size_t)d * HEADS + h];
        atomicAdd(&po[h * DIMS + lane], pl[h * DIMS + lane] * ah);
    }
}

// ---------------------------------------------------------------------------
// out = relu(x + agg + bias)
// ---------------------------------------------------------------------------
__global__ void final_kernel(const float* __restrict__ x,
                             const float* __restrict__ bias,
                             float* __restrict__ out)
{
    const int i = blockIdx.x * blockDim.x + threadIdx.x;
    if (i >= NF) return;
    const float v = x[i] + out[i] + bias[i % FEAT];
    out[i] = (v > 0.0f) ? v : 0.0f;
}

// ---------------------------------------------------------------------------
extern "C" void kernel_launch(void* const* d_in, const int* in_sizes, int n_in,
                              void* d_out, int out_size, void* d_ws, size_t ws_size,
                              hipStream_t stream) {
    const float* x    = (const float*)d_in[0];
    const int*   ei   = (const int*)  d_in[1];
    const float* ew   = (const float*)d_in[2];
    const float* Wl   = (const float*)d_in[3];
    const float* bl   = (const float*)d_in[4];
    const float* Wr   = (const float*)d_in[5];
    const float* br   = (const float*)d_in[6];
    const float* We   = (const float*)d_in[7];
    const float* att  = (const float*)d_in[8];
    const float* bias = (const float*)d_in[9];
    float* out = (float*)d_out;

    // workspace layout (floats): ~90.4 MB total
    float* ws    = (float*)d_ws;
    float* xl    = ws;                                 // NF
    float* xr    = xl + (size_t)NF;                    // NF
    float* alpha = xr + (size_t)NF;                    // E_TOT*HEADS
    float* amax  = alpha + (size_t)E_TOT * HEADS;      // NH
    float* denom = amax + (size_t)NH;                  // NH
    float* wsum  = denom + (size_t)NH;                 // N_NODES
    float* cnt   = wsum + (size_t)N_NODES;             // N_NODES

    // zero: denom, wsum, cnt (contiguous) and the output accumulator
    hipMemsetAsync(denom, 0, (size_t)(NH + 2 * N_NODES) * sizeof(float), stream);
    hipMemsetAsync(out, 0, (size_t)NF * sizeof(float), stream);

    // amax = -inf
    fill_kernel<<<(NH + 255) / 256, 256, 0, stream>>>(amax, NH, -INFINITY);

    // self-loop attr statistics
    deg_kernel<<<(N_EDGES + 255) / 256, 256, 0, stream>>>(ei, ew, wsum, cnt);

    // WMMA GEMMs: x_l, x_r
    gemm_xl_xr_kernel<<<N_NODES / 16, 256, 0, stream>>>(x, Wl, bl, Wr, br, xl, xr);

    // attention scores + segment max
    alpha_kernel<<<E_TOT / 8, 256, 0, stream>>>(ei, ew, xl, xr, We, att,
                                                wsum, cnt, alpha, amax);

    // exp + segment sum
    exp_kernel<<<(E_TOT * HEADS + 255) / 256, 256, 0, stream>>>(ei, alpha, amax, denom);

    // weighted scatter-aggregate
    agg_kernel<<<E_TOT / 8, 256, 0, stream>>>(ei, xl, alpha, denom, out);

    // residual + bias + relu
    final_kernel<<<(NF + 255) / 256, 256, 0, stream>>>(x, bias, out);
}